// Net_7241314861516
// MI455X (gfx1250) — compile-verified
//
#include <hip/hip_runtime.h>

#define NLAYERS 5
#define HID     3
#define NGATE   12      // 4*HID
#define SEQT    512

typedef __attribute__((ext_vector_type(2))) float v2f;
typedef __attribute__((ext_vector_type(8))) float v8f;

// ---- fast transcendentals (gfx1250 has V_TANH_F32, confirmed round 1) -----
__device__ __forceinline__ float frcp(float x) {
#if __has_builtin(__builtin_amdgcn_rcpf)
    return __builtin_amdgcn_rcpf(x);
#else
    return 1.0f / x;
#endif
}

__device__ __forceinline__ float ftanh(float x) {
#if __has_builtin(__builtin_amdgcn_tanhf)
    return __builtin_amdgcn_tanhf(x);
#else
    float e = __expf(2.0f * x);
    return 1.0f - 2.0f * frcp(e + 1.0f);
#endif
}

__device__ __forceinline__ float fsig(float x) {
    return 0.5f + 0.5f * ftanh(0.5f * x);
}

// ---- half-wave swap: lane i <-> lane i+16 ---------------------------------
// Prefer VALU v_permlanex16_b32 (identity nibble selectors) over the LDS-pipe
// ds_bpermute that __shfl_xor lowers to: the swap sits on the recurrence's
// critical path, and permlanex16 avoids DScnt waits entirely.
__device__ __forceinline__ float xswap16(float v) {
#if __has_builtin(__builtin_amdgcn_permlanex16)
    int r = __builtin_amdgcn_permlanex16(__float_as_int(v), __float_as_int(v),
                                         (int)0x76543210, (int)0xFEDCBA98,
                                         false, false);
    return __int_as_float(r);
#else
    return __shfl_xor(v, 16, 32);
#endif
}

// ---- 16x16x4 f32 WMMA (D = A*B + C), guarded with a correct emulation -----
__device__ __forceinline__ v8f wmma4(v2f A, v2f B, v8f C) {
#if __has_builtin(__builtin_amdgcn_wmma_f32_16x16x4_f32)
    return __builtin_amdgcn_wmma_f32_16x16x4_f32(false, A, false, B, (short)0, C,
                                                 false, false);
#else
    const int lane = threadIdx.x & 31;
    const int half = lane >> 4;
    const int n    = lane & 15;
    v8f D;
#pragma unroll
    for (int r = 0; r < 8; ++r) {
        int M = r + 8 * half;
        float acc = C[r];
#pragma unroll
        for (int k = 0; k < 4; ++k) {
            float av = __shfl((k & 1) ? A.y : A.x, (M & 15) + 16 * (k >> 1), 32);
            float bv = __shfl((k & 1) ? B.y : B.x, n + 16 * (k >> 1), 32);
            acc = fmaf(av, bv, acc);
        }
        D[r] = acc;
    }
    return D;
#endif
}

// ---------------------------------------------------------------------------
// One wave = 32 samples (two 16-sample WMMA column groups A and B).
// Gate GEMM per layer-step:  D(12x16) = [Wih|Whh](12x6) x [x;h](6x16) + bias
// via two chained 16x16x4 WMMAs per group (K=3 padded to 4).
// D layout: VGPR r, lane l -> gate row (r + 8*(l>=16)), sample col (l&15):
// low lanes own group-A samples, high lanes own group-B samples, so every
// lane does the full nonlinear cell update for exactly one sample.
// ---------------------------------------------------------------------------
__global__ __launch_bounds__(256, 1)
void lstm5_wmma_kernel(const float* __restrict__ x,
                       const float* __restrict__ Wih,   // (5,12,3)
                       const float* __restrict__ Whh,   // (5,12,3)
                       const float* __restrict__ bih,   // (5,12)
                       const float* __restrict__ bhh,   // (5,12)
                       const float* __restrict__ Wlin,  // (3,3)
                       const float* __restrict__ blin,  // (3,)
                       float* __restrict__ out,         // (B,3)
                       int Btot)
{
    const int lane = threadIdx.x & 31;
    const int wave = threadIdx.x >> 5;
    const int half = lane >> 4;            // 0 = low half, 1 = high half
    const int m    = lane & 15;            // A row index, B/C/D column index
    const int k0   = 2 * half;             // K index held by this half (reg 0)
    const int k1   = k0 + 1;               // K index held by this half (reg 1)

    const int base = (blockIdx.x * 8 + wave) * 32;       // 32 samples / wave
    int myS = base + lane;                                // sample this lane owns
    myS = min(myS, Btot - 1);

    const float* xM = x + (size_t)myS * SEQT * HID;       // own contiguous stream

    // ---- resident weights in A-matrix layout (2 VGPRs per 16x4 matrix) ----
    v2f aih[NLAYERS], ahh[NLAYERS];
    v8f cbias[NLAYERS];
#pragma unroll
    for (int l = 0; l < NLAYERS; ++l) {
        bool v0 = (m < NGATE);                 // k0 in {0,2}, always < HID
        bool v1 = (m < NGATE) && (k1 < HID);   // k1 in {1,3}
        int i0 = l * NGATE * HID + (v0 ? (m * HID + k0) : 0);
        int i1 = l * NGATE * HID + (v1 ? (m * HID + k1) : 0);
        float wi0 = Wih[i0], wi1 = Wih[i1];
        float wh0 = Whh[i0], wh1 = Whh[i1];
        aih[l].x = v0 ? wi0 : 0.0f;  aih[l].y = v1 ? wi1 : 0.0f;
        ahh[l].x = v0 ? wh0 : 0.0f;  ahh[l].y = v1 ? wh1 : 0.0f;
#pragma unroll
        for (int r = 0; r < 8; ++r) {          // bias in C/D layout
            int row = r + 8 * half;
            bool vb = (row < NGATE);
            int ib  = l * NGATE + (vb ? row : 0);
            float bb = bih[ib] + bhh[ib];
            cbias[l][r] = vb ? bb : 0.0f;
        }
    }

    // ---- recurrent state -------------------------------------------------
    float c0[NLAYERS], c1[NLAYERS], c2[NLAYERS];   // cell state of owned sample
    v2f hBA[NLAYERS], hBB[NLAYERS];                // h in B-matrix layout
#pragma unroll
    for (int l = 0; l < NLAYERS; ++l) {
        c0[l] = c1[l] = c2[l] = 0.0f;
        hBA[l].x = hBA[l].y = 0.0f;
        hBB[l].x = hBB[l].y = 0.0f;
    }
    float last0 = 0.0f, last1 = 0.0f, last2 = 0.0f;

    // ---- time loop -------------------------------------------------------
    for (int t = 0; t < SEQT; ++t) {
        if ((t & 7) == 0)
            __builtin_prefetch(xM + (t + 32) * HID, 0, 3);

        // own sample's x_t: one contiguous 12B load per lane
        float x0 = xM[t * HID + 0];
        float x1 = xM[t * HID + 1];
        float x2 = xM[t * HID + 2];
        // distribute into B-matrix layout for both groups via half-swaps:
        //  BinA: K0=compA0(low), K1=compA1(low), K2=compA2(high)
        //  BinB: K0=compB0, K1=compB1 (group-B comps live in high lanes)
        float w2 = xswap16(x2);     // high lane receives group-A comp2
        float w0 = xswap16(x0);     // low lane receives group-B comp0
        float w1 = xswap16(x1);     // low lane receives group-B comp1
        v2f BinA, BinB;
        BinA.x = half ? w2 : x0;  BinA.y = half ? 0.0f : x1;
        BinB.x = half ? x2 : w0;  BinB.y = half ? 0.0f : w1;

#pragma unroll
        for (int l = 0; l < NLAYERS; ++l) {
            // gates = Wih*in + Whh*h + b  (two chained WMMAs per group)
            v8f dA = wmma4(aih[l], BinA,   cbias[l]);
            dA     = wmma4(ahh[l], hBA[l], dA);
            v8f dB = wmma4(aih[l], BinB,   cbias[l]);
            dB     = wmma4(ahh[l], hBB[l], dB);

            // gather the 12 gate rows of this lane's own sample
            float g[NGATE];
#pragma unroll
            for (int r = 0; r < 8; ++r) {
                float sw = xswap16(dB[r]);      // rows 0..7 of group B -> high
                g[r] = half ? sw : dA[r];
            }
#pragma unroll
            for (int r = 0; r < 4; ++r) {
                float sw = xswap16(dA[r]);      // rows 8..11 of group A -> low
                g[8 + r] = half ? dB[r] : sw;
            }

            // LSTM cell update: gate order [i(0:3) f(3:6) g(6:9) o(9:12)]
            c0[l] = fsig(g[3]) * c0[l] + fsig(g[0]) * ftanh(g[6]);
            c1[l] = fsig(g[4]) * c1[l] + fsig(g[1]) * ftanh(g[7]);
            c2[l] = fsig(g[5]) * c2[l] + fsig(g[2]) * ftanh(g[8]);
            float h0f = fsig(g[9])  * ftanh(c0[l]);
            float h1f = fsig(g[10]) * ftanh(c1[l]);
            float h2f = fsig(g[11]) * ftanh(c2[l]);

            // repack fresh h into B layout: recurrent input for t+1 AND input
            // to the next layer at this t
            float s0 = xswap16(h0f);
            float s1 = xswap16(h1f);
            float s2 = xswap16(h2f);
            v2f nA, nB;
            nA.x = half ? s2  : h0f;  nA.y = half ? 0.0f : h1f;
            nB.x = half ? h2f : s0;   nB.y = half ? 0.0f : s1;
            hBA[l] = nA;  hBB[l] = nB;
            BinA   = nA;  BinB   = nB;

            if (l == NLAYERS - 1) { last0 = h0f; last1 = h1f; last2 = h2f; }
        }
    }

    // ---- head: y = max((1 + last @ Wlin^T + blin) * x[:, -1, :], 0) -------
    float xl0 = xM[(SEQT - 1) * HID + 0];
    float xl1 = xM[(SEQT - 1) * HID + 1];
    float xl2 = xM[(SEQT - 1) * HID + 2];
#pragma unroll
    for (int j = 0; j < HID; ++j) {
        float lin = blin[j] + last0 * Wlin[j * HID + 0]
                            + last1 * Wlin[j * HID + 1]
                            + last2 * Wlin[j * HID + 2];
        float xl = (j == 0) ? xl0 : (j == 1) ? xl1 : xl2;
        out[(size_t)myS * HID + j] = fmaxf((1.0f + lin) * xl, 0.0f);
    }
}

extern "C" void kernel_launch(void* const* d_in, const int* in_sizes, int n_in,
                              void* d_out, int out_size, void* d_ws, size_t ws_size,
                              hipStream_t stream) {
    const float* x    = (const float*)d_in[0];
    const float* Wih  = (const float*)d_in[1];
    const float* Whh  = (const float*)d_in[2];
    const float* bihp = (const float*)d_in[3];
    const float* bhhp = (const float*)d_in[4];
    const float* Wlin = (const float*)d_in[5];
    const float* blin = (const float*)d_in[6];
    float* out = (float*)d_out;

    const int Btot   = in_sizes[0] / (SEQT * HID);   // 32768
    const int blocks = (Btot + 255) / 256;           // 256 samples per block
    hipLaunchKernelGGL(lstm5_wmma_kernel, dim3(blocks), dim3(256), 0, stream,
                       x, Wih, Whh, bihp, bhhp, Wlin, blin, out, Btot);
}